// ModelShallow_82592221102830
// MI455X (gfx1250) — compile-verified
//
#include <hip/hip_runtime.h>
#include <math.h>

typedef __attribute__((ext_vector_type(2))) float v2f;
typedef __attribute__((ext_vector_type(8))) float v8f;

// ---------------------------------------------------------------------------
// Degree / normalization
// ---------------------------------------------------------------------------
__global__ __launch_bounds__(256) void k_deg_init(float* deg, int n) {
    int i = blockIdx.x * 256 + threadIdx.x;
    if (i < n) deg[i] = 1.0f;   // self-loop contributes 1
}

__global__ __launch_bounds__(256) void k_deg_scatter(const int* __restrict__ dst,
                                                     float* deg, int e) {
    int i = blockIdx.x * 256 + threadIdx.x;
    if (i < e) atomicAdd(&deg[dst[i]], 1.0f);
}

__global__ __launch_bounds__(256) void k_dinv(const float* __restrict__ deg,
                                              float* dinv, int n) {
    int i = blockIdx.x * 256 + threadIdx.x;
    if (i < n) dinv[i] = rsqrtf(deg[i]);   // deg >= 1 always (self-loops)
}

// ---------------------------------------------------------------------------
// Pad W2 [64,40] -> [64,48] with zeros (unpredicated B loads in WMMA kernel)
// ---------------------------------------------------------------------------
__global__ __launch_bounds__(256) void k_pad_w2(const float* __restrict__ w2,
                                                float* __restrict__ w2p) {
    int i = blockIdx.x * 256 + threadIdx.x;   // 64*48 = 3072 threads
    if (i < 64 * 48) {
        int k = i / 48, c = i % 48;
        w2p[i] = (c < 40) ? w2[k * 40 + c] : 0.0f;
    }
}

// ---------------------------------------------------------------------------
// GEMM1: H[N,64] = X[N,128] @ W[128,64]  (fp32 WMMA 16x16x4)
// One wave per 16x16 output tile. blockDim=128 -> 4 waves -> the 4 N-tiles.
// A layout (16x4): lanes 0-15 hold M=lane, K={k,k+1}; lanes 16-31 K={k+2,k+3}
// B layout (4x16): lanes hold N=lane&15, rows mirror A's K split.
// C/D layout: VGPR v holds row m0 + v + (lane>=16 ? 8 : 0), col n0 + (lane&15)
// ---------------------------------------------------------------------------
__global__ __launch_bounds__(128) void k_gemm1_wmma(const float* __restrict__ X,
                                                    const float* __restrict__ W,
                                                    float* __restrict__ H) {
    const int wave  = threadIdx.x >> 5;
    const int lane  = threadIdx.x & 31;
    const int m0    = blockIdx.x * 16;
    const int n0    = wave * 16;
    const int lr    = lane & 15;
    const int khalf = (lane >> 4) << 1;                 // 0 or 2

    const float* xrow = X + (size_t)(m0 + lr) * 128 + khalf;
    const float* wcol = W + n0 + lr;

    v8f acc = {};
    #pragma unroll
    for (int k = 0; k < 128; k += 4) {
        v2f a, b;
        a.x = xrow[k];
        a.y = xrow[k + 1];
        b.x = wcol[(size_t)(k + khalf) * 64];
        b.y = wcol[(size_t)(k + khalf + 1) * 64];
        acc = __builtin_amdgcn_wmma_f32_16x16x4_f32(false, a, false, b,
                                                    (short)0, acc, false, false);
    }

    float* hp = H + (size_t)(m0 + ((lane >> 4) << 3)) * 64 + n0 + lr;
    #pragma unroll
    for (int v = 0; v < 8; ++v) hp[(size_t)v * 64] = acc[v];
}

// ---------------------------------------------------------------------------
// GEMM2: H[N,48] = A[N,64] @ W[64,48]   (W pre-padded; cols 40..47 are zero)
// blockDim=96 -> 3 waves -> 3 N-tiles covering 48 padded columns.
// ---------------------------------------------------------------------------
__global__ __launch_bounds__(96) void k_gemm2_wmma(const float* __restrict__ A,
                                                   const float* __restrict__ W,
                                                   float* __restrict__ H) {
    const int wave  = threadIdx.x >> 5;
    const int lane  = threadIdx.x & 31;
    const int m0    = blockIdx.x * 16;
    const int n0    = wave * 16;
    const int lr    = lane & 15;
    const int khalf = (lane >> 4) << 1;

    const float* arow = A + (size_t)(m0 + lr) * 64 + khalf;
    const float* wcol = W + n0 + lr;

    v8f acc = {};
    #pragma unroll
    for (int k = 0; k < 64; k += 4) {
        v2f a, b;
        a.x = arow[k];
        a.y = arow[k + 1];
        b.x = wcol[(size_t)(k + khalf) * 48];
        b.y = wcol[(size_t)(k + khalf + 1) * 48];
        acc = __builtin_amdgcn_wmma_f32_16x16x4_f32(false, a, false, b,
                                                    (short)0, acc, false, false);
    }

    float* hp = H + (size_t)(m0 + ((lane >> 4) << 3)) * 48 + n0 + lr;
    #pragma unroll
    for (int v = 0; v < 8; ++v) hp[(size_t)v * 48] = acc[v];
}

// ---------------------------------------------------------------------------
// Layer aggregation: init with self-loop message + bias, then edge scatter.
// ---------------------------------------------------------------------------
__global__ __launch_bounds__(256) void k_init1(const float* __restrict__ h,
                                               const float* __restrict__ dinv,
                                               const float* __restrict__ b,
                                               float* __restrict__ out, int n) {
    int i = blockIdx.x * 256 + threadIdx.x;     // n*64 elems
    if (i < n * 64) {
        int node = i >> 6, c = i & 63;
        float di = dinv[node];
        out[i] = h[i] * di * di + b[c];
    }
}

__global__ __launch_bounds__(256) void k_scatter1(const int* __restrict__ src,
                                                  const int* __restrict__ dst,
                                                  const float* __restrict__ h,
                                                  const float* __restrict__ dinv,
                                                  float* __restrict__ out, int e) {
    long long i = (long long)blockIdx.x * 256 + threadIdx.x;   // e*64 threads
    if (i < (long long)e * 64) {
        int eidx = (int)(i >> 6), c = (int)(i & 63);
        int s = src[eidx], d = dst[eidx];
        float norm = dinv[s] * dinv[d];
        atomicAdd(&out[(size_t)d * 64 + c], h[(size_t)s * 64 + c] * norm);
    }
}

__global__ __launch_bounds__(256) void k_relu(float* x, int n) {
    int i = blockIdx.x * 256 + threadIdx.x;
    if (i < n) x[i] = fmaxf(x[i], 0.0f);
}

__global__ __launch_bounds__(256) void k_init2(const float* __restrict__ h,  // [N,48]
                                               const float* __restrict__ dinv,
                                               const float* __restrict__ b,
                                               float* __restrict__ out, int n) {
    int i = blockIdx.x * 256 + threadIdx.x;     // n*40 elems
    if (i < n * 40) {
        int node = i / 40, c = i % 40;
        float di = dinv[node];
        out[i] = h[(size_t)node * 48 + c] * di * di + b[c];
    }
}

__global__ __launch_bounds__(256) void k_scatter2(const int* __restrict__ src,
                                                  const int* __restrict__ dst,
                                                  const float* __restrict__ h, // [N,48]
                                                  const float* __restrict__ dinv,
                                                  float* __restrict__ out, int e) {
    long long i = (long long)blockIdx.x * 256 + threadIdx.x;   // e*40 threads
    if (i < (long long)e * 40) {
        int eidx = (int)(i / 40), c = (int)(i % 40);
        int s = src[eidx], d = dst[eidx];
        float norm = dinv[s] * dinv[d];
        atomicAdd(&out[(size_t)d * 40 + c], h[(size_t)s * 48 + c] * norm);
    }
}

// ---------------------------------------------------------------------------
// In-place log_softmax over 40 classes; one wave32 per row.
// ---------------------------------------------------------------------------
__global__ __launch_bounds__(256) void k_logsoftmax(float* __restrict__ out, int n) {
    int row  = blockIdx.x * 8 + (threadIdx.x >> 5);
    int lane = threadIdx.x & 31;
    if (row >= n) return;
    float* p = out + (size_t)row * 40;

    float v0 = p[lane];                                   // lanes 0..31 valid
    float v1 = (lane < 8) ? p[lane + 32] : -1e30f;        // lanes 0..7 -> 32..39

    float m = fmaxf(v0, v1);
    #pragma unroll
    for (int off = 16; off > 0; off >>= 1)
        m = fmaxf(m, __shfl_xor(m, off, 32));

    float s = __expf(v0 - m) + ((lane < 8) ? __expf(v1 - m) : 0.0f);
    #pragma unroll
    for (int off = 16; off > 0; off >>= 1)
        s += __shfl_xor(s, off, 32);

    float ls = m + __logf(s);
    p[lane] = v0 - ls;
    if (lane < 8) p[lane + 32] = v1 - ls;
}

// ---------------------------------------------------------------------------
// Host launcher
// ---------------------------------------------------------------------------
extern "C" void kernel_launch(void* const* d_in, const int* in_sizes, int n_in,
                              void* d_out, int out_size, void* d_ws, size_t ws_size,
                              hipStream_t stream) {
    const float* x   = (const float*)d_in[0];
    const int*   ei  = (const int*)d_in[1];
    const float* W1  = (const float*)d_in[2];
    const float* b1  = (const float*)d_in[3];
    const float* W2  = (const float*)d_in[4];
    const float* b2  = (const float*)d_in[5];
    float* out = (float*)d_out;

    const int N = in_sizes[0] / 128;      // 100000
    const int E = in_sizes[1] / 2;        // 1600000
    const int* src = ei;
    const int* dst = ei + E;

    // workspace layout (floats)
    float* ws   = (float*)d_ws;
    float* deg  = ws;                       // N
    float* dinv = deg  + N;                 // N
    float* h1   = dinv + N;                 // N*64
    float* a1   = h1   + (size_t)N * 64;    // N*64
    float* h2   = a1   + (size_t)N * 64;    // N*48
    float* w2p  = h2   + (size_t)N * 48;    // 64*48

    const int B = 256;
    // degree + norm
    k_deg_init<<<(N + B - 1) / B, B, 0, stream>>>(deg, N);
    k_deg_scatter<<<(E + B - 1) / B, B, 0, stream>>>(dst, deg, E);
    k_dinv<<<(N + B - 1) / B, B, 0, stream>>>(deg, dinv, N);
    k_pad_w2<<<(64 * 48 + B - 1) / B, B, 0, stream>>>(W2, w2p);

    // layer 1
    k_gemm1_wmma<<<N / 16, 128, 0, stream>>>(x, W1, h1);
    k_init1<<<(N * 64 + B - 1) / B, B, 0, stream>>>(h1, dinv, b1, a1, N);
    long long w1cnt = (long long)E * 64;
    k_scatter1<<<(int)((w1cnt + B - 1) / B), B, 0, stream>>>(src, dst, h1, dinv, a1, E);
    k_relu<<<(N * 64 + B - 1) / B, B, 0, stream>>>(a1, N * 64);

    // layer 2
    k_gemm2_wmma<<<N / 16, 96, 0, stream>>>(a1, w2p, h2);
    k_init2<<<(N * 40 + B - 1) / B, B, 0, stream>>>(h2, dinv, b2, out, N);
    long long w2cnt = (long long)E * 40;
    k_scatter2<<<(int)((w2cnt + B - 1) / B), B, 0, stream>>>(src, dst, h2, dinv, out, E);

    // log_softmax in place
    k_logsoftmax<<<(N + 7) / 8, 256, 0, stream>>>(out, N);
}